// BayesianSTDPModule_22325240005345
// MI455X (gfx1250) — compile-verified
//
#include <hip/hip_runtime.h>
#include <cmath>

typedef __attribute__((ext_vector_type(2))) float v2f;
typedef __attribute__((ext_vector_type(8))) float v8f;
typedef unsigned int u32x4 __attribute__((ext_vector_type(4)));
typedef int i32x4 __attribute__((ext_vector_type(4)));
typedef int i32x8 __attribute__((ext_vector_type(8)));

#define BB 1024
#define II 2048
#define OO 1024
#define SEQ 32
#define ETA_P 1e-3f
#define ETA_M 1e-3f

// ---------------------------------------------------------------------------
// TDM: issue a 2D tile load Global -> LDS (TENSOR_LOAD_TO_LDS).
// D# packing per CDNA5 ISA: group0 = {flags, lds_addr, global_addr, type=2},
// group1 = {data_size=4B | pad ctrl, tensor dims, tile dims, row stride}.
// pad_enable inserts pad_amount DWORDs after each tile row (bank-conflict-free
// LDS pitch without a strided DMA).
// ---------------------------------------------------------------------------
__device__ __forceinline__ void tdm_load_2d(
    unsigned int lds_off_bytes, const void* gptr,
    unsigned int tile_d0, unsigned int tile_d1,
    unsigned long long stride0_elems,
    unsigned int tens_d0, unsigned int tens_d1,
    unsigned int pad_interval_code, unsigned int pad_amount_m1) {
  unsigned long long ga = (unsigned long long)gptr;
  u32x4 g0;
  g0[0] = 1u;                                            // count=1, user D#
  g0[1] = lds_off_bytes;                                 // lds_addr
  g0[2] = (unsigned int)(ga & 0xFFFFFFFFu);              // global_addr[31:0]
  g0[3] = (unsigned int)((ga >> 32) & 0x1FFFFFFu) | (2u << 30);  // [56:32]|type=2
  i32x8 g1;
  g1[0] = (int)((2u << 16) | (1u << 20) |                // data_size=4B, pad_en
                (pad_interval_code << 22) | (pad_amount_m1 << 25));
  g1[1] = (int)((tens_d0 & 0xFFFFu) << 16);              // tensor_dim0 lo
  g1[2] = (int)(((tens_d0 >> 16) & 0xFFFFu) | ((tens_d1 & 0xFFFFu) << 16));
  g1[3] = (int)(((tens_d1 >> 16) & 0xFFFFu) | (tile_d0 << 16));  // tile_dim0
  g1[4] = (int)(tile_d1 & 0xFFFFu);                      // tile_dim1 (dim2=0)
  g1[5] = (int)(stride0_elems & 0xFFFFFFFFull);          // dim0 stride lo
  g1[6] = (int)((stride0_elems >> 32) & 0xFFFFull);      // dim0 stride hi
  g1[7] = 0;
  i32x4 z4 = {0, 0, 0, 0};
#if defined(__clang_major__) && __clang_major__ >= 23
  i32x8 z8 = {0, 0, 0, 0, 0, 0, 0, 0};
  __builtin_amdgcn_tensor_load_to_lds(g0, g1, z4, z4, z8, 0);
#else
  __builtin_amdgcn_tensor_load_to_lds(g0, g1, z4, z4, 0);
#endif
}

// The TDM writes LDS behind the compiler's back (the LDS address travels as a
// plain integer inside the descriptor). Escape the buffer address into an asm
// with a memory clobber so LLVM cannot prove the buffer is never written and
// fold the ds_loads to undef.
#define LDS_ESCAPE(buf) \
  { void* _q = (void*)(buf); asm volatile("" : : "v"(_q) : "memory"); }
#define DMA_FENCE() asm volatile("" ::: "memory")

// ---------------------------------------------------------------------------
// Init: copy weights into live w buffer (d_out w-region), v=0, tpost=1.
// ---------------------------------------------------------------------------
__global__ __launch_bounds__(256) void snn_init_kernel(
    const float* __restrict__ w_in, float* __restrict__ wbuf,
    float* __restrict__ vbuf, float* __restrict__ tpostbuf) {
  size_t idx = (size_t)blockIdx.x * 256 + threadIdx.x;
  wbuf[idx] = w_in[idx];
  if (idx < (size_t)BB * OO) {
    vbuf[idx] = 0.0f;
    tpostbuf[idx] = 1.0f;
  }
}

// ---------------------------------------------------------------------------
// Forward GEMM z_in = S @ W^T + bias, fused LIF + tpost trace update.
// Block tile 128(M=batch) x 64(N=out); 8 waves, each 16x64 via 4 accumulators.
// K chunks of 32 staged into LDS by TDM, double-buffered (wave0 drives DMA).
// LDS pitch 36 floats (32 + 4 pad) -> conflict-free ds_load_b64 fragments.
// ---------------------------------------------------------------------------
#define FW_PITCH 36
#define FW_BUF   ((128 + 64) * FW_PITCH)          // floats per buffer
__global__ __launch_bounds__(256) void snn_fwd_lif_kernel(
    const float* __restrict__ S, const float* __restrict__ W,
    const float* __restrict__ bias, float* __restrict__ zbuf,
    float* __restrict__ vbuf, float* __restrict__ tpostbuf,
    float vdecay, float postdecay) {
  __shared__ float sb[2 * FW_BUF];
  LDS_ESCAPE(sb);

  const int lane = threadIdx.x & 31;
  const int wave = threadIdx.x >> 5;   // 0..7
  const int half = lane >> 4;          // 0/1
  const int r    = lane & 15;
  const int mB = blockIdx.x * 128;     // block batch-row base
  const int o0 = blockIdx.y * 64;      // block output-col base
  const int m0 = mB + wave * 16;

  const int NC = II / 32;              // 64 K-chunks

  // prologue: chunk 0 -> buffer 0
  if (threadIdx.x < 32) {
    tdm_load_2d(0u, S + (size_t)mB * II, 32u, 128u, (unsigned long long)II,
                (unsigned int)II, (unsigned int)BB, 4u, 3u);
    tdm_load_2d((unsigned int)(128 * FW_PITCH * 4), W + (size_t)o0 * II,
                32u, 64u, (unsigned long long)II,
                (unsigned int)II, (unsigned int)OO, 4u, 3u);
  }

  v8f c0 = {}, c1 = {}, c2 = {}, c3 = {};
  for (int kc = 0; kc < NC; ++kc) {
    const int cur = kc & 1;
    if (threadIdx.x < 32) {
      if (kc + 1 < NC) {  // prefetch next chunk into other buffer
        const int nxt = cur ^ 1;
        const int k0n = (kc + 1) * 32;
        tdm_load_2d((unsigned int)(nxt * FW_BUF * 4),
                    S + (size_t)mB * II + k0n, 32u, 128u,
                    (unsigned long long)II, (unsigned int)II,
                    (unsigned int)BB, 4u, 3u);
        tdm_load_2d((unsigned int)((nxt * FW_BUF + 128 * FW_PITCH) * 4),
                    W + (size_t)o0 * II + k0n, 32u, 64u,
                    (unsigned long long)II, (unsigned int)II,
                    (unsigned int)OO, 4u, 3u);
        __builtin_amdgcn_s_wait_tensorcnt(2);  // chunk kc complete
      } else {
        __builtin_amdgcn_s_wait_tensorcnt(0);
      }
    }
    __syncthreads();
    DMA_FENCE();  // DMA-produced LDS data: block load hoisting/folding

    const float* A = &sb[cur * FW_BUF];
    const float* Bt = A + 128 * FW_PITCH;
    const int ar = (wave * 16 + r) * FW_PITCH + 2 * half;
#pragma unroll
    for (int kk = 0; kk < 32; kk += 4) {
      v2f a  = *(const v2f*)&A[ar + kk];
      v2f b0 = *(const v2f*)&Bt[(0  + r) * FW_PITCH + 2 * half + kk];
      v2f b1 = *(const v2f*)&Bt[(16 + r) * FW_PITCH + 2 * half + kk];
      v2f b2 = *(const v2f*)&Bt[(32 + r) * FW_PITCH + 2 * half + kk];
      v2f b3 = *(const v2f*)&Bt[(48 + r) * FW_PITCH + 2 * half + kk];
      c0 = __builtin_amdgcn_wmma_f32_16x16x4_f32(false, a, false, b0, (short)0, c0, false, false);
      c1 = __builtin_amdgcn_wmma_f32_16x16x4_f32(false, a, false, b1, (short)0, c1, false, false);
      c2 = __builtin_amdgcn_wmma_f32_16x16x4_f32(false, a, false, b2, (short)0, c2, false, false);
      c3 = __builtin_amdgcn_wmma_f32_16x16x4_f32(false, a, false, b3, (short)0, c3, false, false);
    }
    __syncthreads();  // buffer consumed; safe to overwrite next iteration
  }

  v8f cc[4] = {c0, c1, c2, c3};
#pragma unroll
  for (int j = 0; j < 4; ++j) {
    const int o = o0 + 16 * j + r;
    const float bi = bias[o];
#pragma unroll
    for (int e = 0; e < 8; ++e) {
      const int b = m0 + e + 8 * half;          // C layout: M = e + 8*half
      const size_t idx = (size_t)b * OO + o;
      const float zin = cc[j][e] + bi;
      const float vm  = vbuf[idx] * vdecay + zin;
      const float z   = (vm >= 1.0f) ? 1.0f : 0.0f;
      vbuf[idx]     = vm * (1.0f - z);
      zbuf[idx]     = z;
      tpostbuf[idx] = tpostbuf[idx] * postdecay + z;  // updated trace for dw
    }
  }
}

// ---------------------------------------------------------------------------
// Fused STDP update: dw[o,i] = sum_b ( eta_p*z[b,o]*(At+Bt*S[b,i])
//                                     - eta_m*tpost[b,o]*S[b,i] ),
// then w = clip(w + dw). tpre eliminated via closed form (spikes constant).
// Block tile 128(O) x 64(I); K chunks of 16 over batch; TDM stages z, tpost
// and S tiles (S shared by both terms), double-buffered.
// Pitches: z/tpost 136 (128+8), S 72 (64+8) -> conflict-free b32 fragments.
// ---------------------------------------------------------------------------
#define ZT_PITCH 136
#define SS_PITCH 72
#define UP_ZOFF  0
#define UP_TOFF  (16 * ZT_PITCH)
#define UP_SOFF  (2 * 16 * ZT_PITCH)
#define UP_BUF   (2 * 16 * ZT_PITCH + 16 * SS_PITCH)   // floats per buffer
__global__ __launch_bounds__(256) void snn_stdp_update_kernel(
    const float* __restrict__ S, const float* __restrict__ zbuf,
    const float* __restrict__ tpostbuf, float* __restrict__ W,
    float At, float Bt) {
  __shared__ float sb[2 * UP_BUF];
  LDS_ESCAPE(sb);

  const int lane = threadIdx.x & 31;
  const int wave = threadIdx.x >> 5;
  const int half = lane >> 4;
  const int r    = lane & 15;
  const int oB = blockIdx.x * 128;
  const int iB = blockIdx.y * 64;
  const int wo = wave * 16 + r;        // this wave's O-column inside z/t tiles

  const int NC = BB / 16;              // 64 K-chunks over batch

  if (threadIdx.x < 32) {
    tdm_load_2d((unsigned int)(UP_ZOFF * 4), zbuf + oB, 128u, 16u,
                (unsigned long long)OO, (unsigned int)OO, (unsigned int)BB, 6u, 7u);
    tdm_load_2d((unsigned int)(UP_TOFF * 4), tpostbuf + oB, 128u, 16u,
                (unsigned long long)OO, (unsigned int)OO, (unsigned int)BB, 6u, 7u);
    tdm_load_2d((unsigned int)(UP_SOFF * 4), S + iB, 64u, 16u,
                (unsigned long long)II, (unsigned int)II, (unsigned int)BB, 5u, 7u);
  }

  v8f c0 = {}, c1 = {}, c2 = {}, c3 = {};
  for (int kc = 0; kc < NC; ++kc) {
    const int cur = kc & 1;
    if (threadIdx.x < 32) {
      if (kc + 1 < NC) {
        const int nxt = cur ^ 1;
        const size_t k0n = (size_t)(kc + 1) * 16;
        tdm_load_2d((unsigned int)((nxt * UP_BUF + UP_ZOFF) * 4),
                    zbuf + k0n * OO + oB, 128u, 16u,
                    (unsigned long long)OO, (unsigned int)OO, (unsigned int)BB, 6u, 7u);
        tdm_load_2d((unsigned int)((nxt * UP_BUF + UP_TOFF) * 4),
                    tpostbuf + k0n * OO + oB, 128u, 16u,
                    (unsigned long long)OO, (unsigned int)OO, (unsigned int)BB, 6u, 7u);
        tdm_load_2d((unsigned int)((nxt * UP_BUF + UP_SOFF) * 4),
                    S + k0n * II + iB, 64u, 16u,
                    (unsigned long long)II, (unsigned int)II, (unsigned int)BB, 5u, 7u);
        __builtin_amdgcn_s_wait_tensorcnt(3);  // chunk kc complete
      } else {
        __builtin_amdgcn_s_wait_tensorcnt(0);
      }
    }
    __syncthreads();
    DMA_FENCE();

    const float* zt = &sb[cur * UP_BUF + UP_ZOFF];
    const float* tt = &sb[cur * UP_BUF + UP_TOFF];
    const float* st = &sb[cur * UP_BUF + UP_SOFF];
#pragma unroll
    for (int kk = 0; kk < 16; kk += 4) {
      const int ka = kk + 2 * half;
      v2f a1, a2;
      a1.x =  ETA_P * zt[ka * ZT_PITCH + wo];
      a1.y =  ETA_P * zt[(ka + 1) * ZT_PITCH + wo];
      a2.x = -ETA_M * tt[ka * ZT_PITCH + wo];
      a2.y = -ETA_M * tt[(ka + 1) * ZT_PITCH + wo];
      v2f s0, s1, s2, s3;
      s0.x = st[ka * SS_PITCH +  0 + r]; s0.y = st[(ka + 1) * SS_PITCH +  0 + r];
      s1.x = st[ka * SS_PITCH + 16 + r]; s1.y = st[(ka + 1) * SS_PITCH + 16 + r];
      s2.x = st[ka * SS_PITCH + 32 + r]; s2.y = st[(ka + 1) * SS_PITCH + 32 + r];
      s3.x = st[ka * SS_PITCH + 48 + r]; s3.y = st[(ka + 1) * SS_PITCH + 48 + r];
      v2f p0 = {At + Bt * s0.x, At + Bt * s0.y};
      v2f p1 = {At + Bt * s1.x, At + Bt * s1.y};
      v2f p2 = {At + Bt * s2.x, At + Bt * s2.y};
      v2f p3 = {At + Bt * s3.x, At + Bt * s3.y};
      c0 = __builtin_amdgcn_wmma_f32_16x16x4_f32(false, a1, false, p0, (short)0, c0, false, false);
      c1 = __builtin_amdgcn_wmma_f32_16x16x4_f32(false, a1, false, p1, (short)0, c1, false, false);
      c2 = __builtin_amdgcn_wmma_f32_16x16x4_f32(false, a1, false, p2, (short)0, c2, false, false);
      c3 = __builtin_amdgcn_wmma_f32_16x16x4_f32(false, a1, false, p3, (short)0, c3, false, false);
      c0 = __builtin_amdgcn_wmma_f32_16x16x4_f32(false, a2, false, s0, (short)0, c0, false, false);
      c1 = __builtin_amdgcn_wmma_f32_16x16x4_f32(false, a2, false, s1, (short)0, c1, false, false);
      c2 = __builtin_amdgcn_wmma_f32_16x16x4_f32(false, a2, false, s2, (short)0, c2, false, false);
      c3 = __builtin_amdgcn_wmma_f32_16x16x4_f32(false, a2, false, s3, (short)0, c3, false, false);
    }
    __syncthreads();
  }

  v8f cc[4] = {c0, c1, c2, c3};
#pragma unroll
  for (int j = 0; j < 4; ++j) {
#pragma unroll
    for (int e = 0; e < 8; ++e) {
      const int o = oB + wave * 16 + e + 8 * half;
      const int i = iB + 16 * j + r;
      const size_t idx = (size_t)o * II + i;
      float w = W[idx] + cc[j][e];
      w = fminf(1.0f, fmaxf(-1.0f, w));
      W[idx] = w;
    }
  }
}

// ---------------------------------------------------------------------------
// Host driver: init, then 32 x (forward+LIF, stdp-update) on the stream.
// State lives in d_out (z, v, w regions) + tpost in d_ws (4 MB).
// ---------------------------------------------------------------------------
extern "C" void kernel_launch(void* const* d_in, const int* in_sizes, int n_in,
                              void* d_out, int out_size, void* d_ws, size_t ws_size,
                              hipStream_t stream) {
  (void)in_sizes; (void)n_in; (void)out_size; (void)ws_size;
  const float* S    = (const float*)d_in[0];
  const float* Win  = (const float*)d_in[1];
  const float* bias = (const float*)d_in[2];
  float* out   = (float*)d_out;
  float* zbuf  = out;                               // [B,O]
  float* vbuf  = out + (size_t)BB * OO;             // [B,O]
  float* wbuf  = out + 2 * (size_t)BB * OO;         // [O,I] live weights
  float* tpost = (float*)d_ws;                      // [B,O]

  snn_init_kernel<<<(unsigned)(((size_t)OO * II) / 256), 256, 0, stream>>>(
      Win, wbuf, vbuf, tpost);

  const float dpre = expf(-0.05f);  // -DT/TAU_PRE = -DT/TAU_POST
  const float vd   = expf(-0.05f);  // -DT/TAU_MEM
  float At = 1.0f;
  for (int s = 0; s < SEQ; ++s) {
    snn_fwd_lif_kernel<<<dim3(BB / 128, OO / 64), 256, 0, stream>>>(
        S, wbuf, bias, zbuf, vbuf, tpost, vd, dpre);
    At *= dpre;  // decay_pre^(s+1): tpre_t = At + Bt * spikes
    const float Bt = (1.0f - At) / (1.0f - dpre);
    snn_stdp_update_kernel<<<dim3(OO / 128, II / 64), 256, 0, stream>>>(
        S, zbuf, tpost, wbuf, At, Bt);
  }
}